// SoftDTWLoss_5652176962170
// MI455X (gfx1250) — compile-verified
//
#include <hip/hip_runtime.h>

// ---------------------------------------------------------------------------
// mean over B of per-row L2 norms:  out = (1/B) * sum_b sqrt( sum_t (yp-yt)^2 )
// B = 64, T = 524288.  Pure streaming reduction, HBM-bound (256 MiB @ 23.3TB/s).
// ---------------------------------------------------------------------------

#define ROWS            64
#define ROW_LEN         524288
#define CHUNKS_PER_ROW  128
#define BLOCK_THREADS   256
#define CHUNK_ELEMS     (ROW_LEN / CHUNKS_PER_ROW)   // 4096 floats
#define VECS_PER_THREAD (CHUNK_ELEMS / (BLOCK_THREADS * 4))  // 4 x float4

typedef __attribute__((ext_vector_type(4))) float f4;
typedef __attribute__((ext_vector_type(2))) float v2f;
typedef __attribute__((ext_vector_type(8))) float v8f;

__global__ __launch_bounds__(BLOCK_THREADS)
void sumsq_partial_kernel(const float* __restrict__ yp,
                          const float* __restrict__ yt,
                          float* __restrict__ partials) {
    const int chunk = blockIdx.x;      // 0..127
    const int row   = blockIdx.y;      // 0..63
    const size_t base = (size_t)row * ROW_LEN + (size_t)chunk * CHUNK_ELEMS;

    const f4* __restrict__ p4 = (const f4*)(yp + base);
    const f4* __restrict__ t4 = (const f4*)(yt + base);

    // Per-thread sum of squared diffs: 4 x b128 loads per array, NT hint
    // (one-shot 256 MiB stream -> don't pollute WGP$/L2).
    float acc = 0.0f;
#pragma unroll
    for (int i = 0; i < VECS_PER_THREAD; ++i) {
        const int idx = i * BLOCK_THREADS + threadIdx.x;
        f4 a = __builtin_nontemporal_load(p4 + idx);
        f4 b = __builtin_nontemporal_load(t4 + idx);
        float dx = a.x - b.x;
        float dy = a.y - b.y;
        float dz = a.z - b.z;
        float dw = a.w - b.w;
        acc = fmaf(dx, dx, acc);
        acc = fmaf(dy, dy, acc);
        acc = fmaf(dz, dz, acc);
        acc = fmaf(dw, dw, acc);
    }

    // Block reduction of 256 partials via the matrix pipe:
    // stage partials in LDS, then wave 0 runs a chain of 4x
    // v_wmma_f32_16x16x4_f32 with A == ones(16x4).  With A==1,
    // D[m][n] = sum_k B[k][n] (+C), so summing one D-row's 16 columns yields
    // the total of all 64 packed B values -- independent of B's lane layout.
    __shared__ float red[BLOCK_THREADS];
    red[threadIdx.x] = acc;
    __syncthreads();

    if (threadIdx.x < 32) {            // exactly wave 0 -> EXEC all ones
        v2f ones;
        ones[0] = 1.0f;
        ones[1] = 1.0f;
        v8f c = {};
#pragma unroll
        for (int ch = 0; ch < 4; ++ch) {
            v2f b;
            b[0] = red[ch * 64 + threadIdx.x];
            b[1] = red[ch * 64 + 32 + threadIdx.x];
            c = __builtin_amdgcn_wmma_f32_16x16x4_f32(
                    /*neg_a=*/false, ones,
                    /*neg_b=*/false, b,
                    /*c_mod=*/(short)0, c,
                    /*reuse_a=*/false, /*reuse_b=*/false);
        }
        // c[0]: lane L holds column-sum for N = L % 16.  Sum N=0..15 within
        // the low 16-lane group -> lane 0 holds the full 256-value total.
        float v = c[0];
        v += __shfl_xor(v, 1, 32);
        v += __shfl_xor(v, 2, 32);
        v += __shfl_xor(v, 4, 32);
        v += __shfl_xor(v, 8, 32);
        if (threadIdx.x == 0) {
            partials[row * CHUNKS_PER_ROW + chunk] = v;
        }
    }
}

// Single wave32: deterministic finalize.  For each row, sum its 128 chunk
// partials (fixed order), sqrt, accumulate, then write mean.
__global__ __launch_bounds__(32)
void finalize_kernel(const float* __restrict__ partials,
                     float* __restrict__ out) {
    const int lane = threadIdx.x;
    float acc = 0.0f;
    for (int r = 0; r < ROWS; ++r) {
        const float* p = partials + r * CHUNKS_PER_ROW;
        float s = p[lane] + p[lane + 32] + p[lane + 64] + p[lane + 96];
        s += __shfl_xor(s, 16, 32);
        s += __shfl_xor(s, 8, 32);
        s += __shfl_xor(s, 4, 32);
        s += __shfl_xor(s, 2, 32);
        s += __shfl_xor(s, 1, 32);
        acc += sqrtf(s);               // all lanes hold identical s
    }
    if (lane == 0) {
        out[0] = acc * (1.0f / (float)ROWS);
    }
}

extern "C" void kernel_launch(void* const* d_in, const int* in_sizes, int n_in,
                              void* d_out, int out_size, void* d_ws, size_t ws_size,
                              hipStream_t stream) {
    const float* yp = (const float*)d_in[0];
    const float* yt = (const float*)d_in[1];
    float* partials = (float*)d_ws;    // ROWS * CHUNKS_PER_ROW floats = 32 KiB
    float* out      = (float*)d_out;

    dim3 grid(CHUNKS_PER_ROW, ROWS);   // 128 x 64 = 8192 blocks, 8 waves each
    sumsq_partial_kernel<<<grid, dim3(BLOCK_THREADS), 0, stream>>>(yp, yt, partials);
    finalize_kernel<<<1, 32, 0, stream>>>(partials, out);
}